// SAGE_61495341744416
// MI455X (gfx1250) — compile-verified
//
#include <hip/hip_runtime.h>

// ---------------------------------------------------------------------------
// GraphSAGE 2-layer forward for MI455X (gfx1250, wave32).
//  - Edge aggregation: edge-parallel float4 gathers + global_atomic_add_f32.
//    All hot buffers (x 12.8MB, agg 12.8MB, h1/agg2 6.4MB) fit in 192MB L2,
//    so the atomics/gathers run at L2 speed.
//  - Dense part: fused [mean|root] @ [Wl;Wr] GEMM using V_WMMA_F32_16X16X4_F32
//    (exact f32 matrix math; GEMM is ~0.8 GFLOP total, far from bottleneck).
//  - Weights staged in LDS as K-pair-interleaved float2 so each B fragment is
//    a single aligned ds_load_b64 into an even VGPR pair (no repack movs).
// ---------------------------------------------------------------------------

typedef float v2f __attribute__((ext_vector_type(2)));
typedef float v8f __attribute__((ext_vector_type(8)));

#define N_IN_F  64
#define N_HID   32
#define N_OUT_F 64

__global__ void zero_kernel(float* __restrict__ p, long long n) {
  long long i = (long long)blockIdx.x * blockDim.x + threadIdx.x;
  if (i < n) p[i] = 0.0f;
}

// One edge is handled by F/4 threads; each thread gathers a float4 from the
// source row and atomically accumulates 4 floats into the destination row.
template <int F, bool COUNT_DEG>
__global__ void edge_agg_kernel(const float* __restrict__ feat,
                                const int* __restrict__ src,
                                const int* __restrict__ dst,
                                float* __restrict__ agg,
                                float* __restrict__ deg,
                                int n_edges) {
  constexpr int CH = F / 4;  // chunks per edge (power of two)
  long long t = (long long)blockIdx.x * blockDim.x + threadIdx.x;
  long long e = t / CH;
  int c = (int)(t % CH);
  if (e >= n_edges) return;
  int s = src[e];
  int d = dst[e];
  const float4 v =
      *reinterpret_cast<const float4*>(feat + (size_t)s * F + (size_t)c * 4);
  float* a = agg + (size_t)d * F + (size_t)c * 4;
  atomicAdd(a + 0, v.x);
  atomicAdd(a + 1, v.y);
  atomicAdd(a + 2, v.z);
  atomicAdd(a + 3, v.w);
  if (COUNT_DEG && c == 0) atomicAdd(deg + d, 1.0f);
}

// Fused SAGE layer: out = relu([agg/deg | root] @ [Wl ; Wr] + bias)
// Each wave computes a 16-row output tile with V_WMMA_F32_16X16X4_F32.
// FIN = input features (64 or 32); NOUT = output features (32 or 64).
template <int FIN, int NOUT>
__global__ void sage_layer_kernel(const float* __restrict__ agg,
                                  const float* __restrict__ deg,
                                  const float* __restrict__ root,
                                  const float* __restrict__ Wl,
                                  const float* __restrict__ Wr,
                                  const float* __restrict__ bias,
                                  float* __restrict__ out,
                                  int n_nodes) {
  constexpr int KCAT = 2 * FIN;      // concatenated K dimension
  constexpr int NT   = NOUT / 16;    // 16-wide N tiles
  constexpr int KP   = KCAT / 2;     // K pairs

  // sWp[k2*NOUT + n] = { Wcat[2*k2][n], Wcat[2*k2+1][n] }  (Wcat = [Wl;Wr])
  __shared__ v2f  sWp[KP * NOUT];
  __shared__ float sB[NOUT];

  for (int i = threadIdx.x; i < KP * NOUT; i += blockDim.x) {
    const int k2 = i / NOUT;
    const int n  = i - k2 * NOUT;
    const int k  = 2 * k2;
    v2f w;
    if (k < FIN) {
      w.x = Wl[(size_t)k * NOUT + n];
      w.y = Wl[(size_t)(k + 1) * NOUT + n];
    } else {
      w.x = Wr[(size_t)(k - FIN) * NOUT + n];
      w.y = Wr[(size_t)(k + 1 - FIN) * NOUT + n];
    }
    sWp[i] = w;
  }
  for (int i = threadIdx.x; i < NOUT; i += blockDim.x) sB[i] = bias[i];
  __syncthreads();

  const int lane  = threadIdx.x & 31;
  const int wave  = threadIdx.x >> 5;
  const int waves = blockDim.x >> 5;
  const int row_base = (blockIdx.x * waves + wave) * 16;
  if (row_base >= n_nodes) return;   // wave-uniform: EXEC stays all-ones below

  // A-fragment lane mapping for 16x4 f32: M = lane&15, K = (lane>>4)*2 + j
  const int m  = lane & 15;
  const int hi = lane >> 4;          // lane half: 0 or 1
  const int kh = hi << 1;            // K offset within 4-wide step: 0 or 2
  int row = row_base + m;
  if (row >= n_nodes) row = n_nodes - 1;  // clamp (stores guarded below)

  const float dg  = deg[row];
  const float inv = 1.0f / fmaxf(dg, 1.0f);

  const float* aggrow  = agg  + (size_t)row * FIN;
  const float* rootrow = root + (size_t)row * FIN;

  v8f acc[NT];
#pragma unroll
  for (int nt = 0; nt < NT; ++nt) acc[nt] = (v8f){0, 0, 0, 0, 0, 0, 0, 0};

#pragma unroll
  for (int k = 0; k < KCAT; k += 4) {  // wave-uniform k; kh in {0,2}
    v2f a;
    if (k < FIN) {  // FIN is a multiple of 4, so no half/half straddle
      a.x = aggrow[k + kh] * inv;
      a.y = aggrow[k + kh + 1] * inv;
    } else {
      a.x = rootrow[k + kh - FIN];
      a.y = rootrow[k + kh + 1 - FIN];
    }
    const int k2 = (k >> 1) + hi;    // pair index for (k+kh, k+kh+1)
#pragma unroll
    for (int nt = 0; nt < NT; ++nt) {
      // B-fragment 4x16 f32: N = lane&15, K = (lane>>4)*2 + j
      const v2f b = sWp[k2 * NOUT + nt * 16 + m];  // one aligned ds_load_b64
      acc[nt] = __builtin_amdgcn_wmma_f32_16x16x4_f32(
          false, a, false, b, (short)0, acc[nt], false, false);
    }
  }

  // D layout: VGPR j -> M = j + (lane>>4)*8, N = lane&15
  const int mh = hi << 3;
#pragma unroll
  for (int nt = 0; nt < NT; ++nt) {
#pragma unroll
    for (int j = 0; j < 8; ++j) {
      const int r = row_base + j + mh;
      if (r < n_nodes) {
        const int n = nt * 16 + m;
        float v = acc[nt][j] + sB[n];
        out[(size_t)r * NOUT + n] = fmaxf(v, 0.0f);
      }
    }
  }
}

extern "C" void kernel_launch(void* const* d_in, const int* in_sizes, int n_in,
                              void* d_out, int out_size, void* d_ws,
                              size_t ws_size, hipStream_t stream) {
  const float* x   = (const float*)d_in[0];
  const int*   ei  = (const int*)d_in[1];
  const float* W1l = (const float*)d_in[2];
  const float* b1  = (const float*)d_in[3];
  const float* W1r = (const float*)d_in[4];
  const float* W2l = (const float*)d_in[5];
  const float* b2  = (const float*)d_in[6];
  const float* W2r = (const float*)d_in[7];
  float* out = (float*)d_out;

  const int n_nodes = in_sizes[0] / N_IN_F;
  const int n_edges = in_sizes[1] / 2;
  const int* src = ei;             // edge_index row 0
  const int* dst = ei + n_edges;   // edge_index row 1

  // Workspace layout (floats): [agg1 n*64][deg n][agg2 n*32][h1 n*32]
  float* ws   = (float*)d_ws;
  float* agg1 = ws;
  float* deg  = agg1 + (size_t)n_nodes * N_IN_F;
  float* agg2 = deg + n_nodes;
  float* h1   = agg2 + (size_t)n_nodes * N_HID;

  // Zero the accumulated region (agg1, deg, agg2 are contiguous).
  long long nz = (long long)n_nodes * (N_IN_F + 1 + N_HID);
  zero_kernel<<<(unsigned)((nz + 255) / 256), 256, 0, stream>>>(ws, nz);

  // Layer 1 aggregation: x[src] -> agg1[dst], plus degree counts.
  long long t1 = (long long)n_edges * (N_IN_F / 4);
  edge_agg_kernel<N_IN_F, true>
      <<<(unsigned)((t1 + 255) / 256), 256, 0, stream>>>(x, src, dst, agg1,
                                                         deg, n_edges);

  // Layer 1 dense: h1 = relu([agg1/deg | x] @ [W1l;W1r] + b1)
  const int rows_per_block = (256 / 32) * 16;  // 8 waves * 16 rows
  const int grid_l = (n_nodes + rows_per_block - 1) / rows_per_block;
  sage_layer_kernel<N_IN_F, N_HID>
      <<<grid_l, 256, 0, stream>>>(agg1, deg, x, W1l, W1r, b1, h1, n_nodes);

  // Layer 2 aggregation: h1[src] -> agg2[dst] (same degrees as layer 1).
  long long t2 = (long long)n_edges * (N_HID / 4);
  edge_agg_kernel<N_HID, false>
      <<<(unsigned)((t2 + 255) / 256), 256, 0, stream>>>(h1, src, dst, agg2,
                                                         deg, n_edges);

  // Layer 2 dense: out = relu([agg2/deg | h1] @ [W2l;W2r] + b2)
  sage_layer_kernel<N_HID, N_OUT_F>
      <<<grid_l, 256, 0, stream>>>(agg2, deg, h1, W2l, W2r, b2, out, n_nodes);
}